// EuclideanCodebook_22419729285665
// MI455X (gfx1250) — compile-verified
//
#include <hip/hip_runtime.h>

typedef __attribute__((ext_vector_type(2))) float v2f;
typedef __attribute__((ext_vector_type(8))) float v8f;
typedef __attribute__((ext_vector_type(4))) int   v4i;

#define DIM            64
#define KSTEPS         16          // 64 / 4 per-WMMA K
#define TILE_N         32          // codewords per iteration (2 x 16 sub-tiles)
#define SUB_N          16
#define ROWS_PER_WAVE  16
#define WAVES_PER_BLK  8
#define ROWS_PER_BLK   (ROWS_PER_WAVE * WAVES_PER_BLK)   // 128
#define LDS_STRIDE     68          // 64 + 4 pad floats -> conflict-free ds_load_b64

// Use CDNA5 async global->LDS DMA (ASYNCcnt) if the toolchain exposes it.
#if defined(__has_builtin)
#if __has_builtin(__builtin_amdgcn_global_load_async_to_lds_b128) && \
    __has_builtin(__builtin_amdgcn_s_wait_asynccnt)
#define USE_ASYNC_LDS 1
#endif
#endif
#ifndef USE_ASYNC_LDS
#define USE_ASYNC_LDS 0
#endif

#if USE_ASYNC_LDS
#define NBUF 2
typedef __attribute__((address_space(1))) v4i* global_v4i_p;
typedef __attribute__((address_space(3))) v4i* lds_v4i_p;
__device__ __forceinline__ void async_tile_load_b128(const float* g, float* l) {
    __builtin_amdgcn_global_load_async_to_lds_b128(
        (global_v4i_p)(void*)g,   // addrspacecast generic -> global
        (lds_v4i_p)(void*)l,      // addrspacecast generic -> LDS
        /*offset=*/0, /*cpol=*/0);
}
#else
#define NBUF 1
#endif

// ---------------------------------------------------------------------------
// Kernel 1: codebook squared norms  esq[k] = sum_d embed[k][d]^2
// ---------------------------------------------------------------------------
__global__ void vq_esq_kernel(const float* __restrict__ embed,
                              float* __restrict__ esq, int K) {
    int k = blockIdx.x * blockDim.x + threadIdx.x;
    if (k >= K) return;
    const float4* row = (const float4*)(embed + (size_t)k * DIM);
    float s = 0.f;
#pragma unroll
    for (int i = 0; i < DIM / 4; ++i) {
        float4 v = row[i];
        s += v.x * v.x + v.y * v.y + v.z * v.z + v.w * v.w;
    }
    esq[k] = s;
}

// ---------------------------------------------------------------------------
// Kernel 2: fused  dist = x @ embed^T - 0.5*esq  -> argmax -> gather
// Per iteration a wave computes a 16(row) x 32(col) distance tile as TWO
// independent 16x16 WMMA accumulator chains (hides v_wmma latency), each
// accumulating K=64 with 16 chained v_wmma_f32_16x16x4_f32. Embed tiles are
// staged in LDS via double-buffered async-to-LDS DMA (ASYNCcnt) when
// available, else a software-pipelined copy.
// ---------------------------------------------------------------------------
__launch_bounds__(WAVES_PER_BLK * 32)
__global__ void vq_argmax_kernel(const float* __restrict__ x,
                                 const float* __restrict__ embed,
                                 const float* __restrict__ esq,
                                 float* __restrict__ out_q,
                                 float* __restrict__ out_idx,
                                 int n_codes) {
    __shared__ float lds[NBUF][TILE_N * LDS_STRIDE];

    const int tid  = threadIdx.x;
    const int wave = tid >> 5;
    const int lane = tid & 31;
    const int l16  = lane & 15;       // row (A) / col (B,C) within sub-tile
    const int hi   = lane >> 4;       // lane-half selects K pair {0,1} vs {2,3}
    const int r0   = blockIdx.x * ROWS_PER_BLK + wave * ROWS_PER_WAVE;

    // ---- preload all A fragments for this wave's 16 rows (K = 0..63) ----
    // ISA layout 32-bit A 16x4: lanes 0-15 M=0..15 {K=0,K=1}, lanes 16-31 {K=2,K=3}
    v2f a[KSTEPS];
    const float* arow = x + (size_t)(r0 + l16) * DIM + hi * 2;
#pragma unroll
    for (int s = 0; s < KSTEPS; ++s)
        a[s] = *(const v2f*)(arow + 4 * s);

    float best[8];
    int   bidx[8];
#pragma unroll
    for (int i = 0; i < 8; ++i) { best[i] = -3.402823466e38f; bidx[i] = 0; }

    // Cooperative staging: tile = 32 codebook rows x 64 floats = 512 float4
    // slots; thread t handles slots t and t+256 (rows srow and srow+16).
    const int srow = tid >> 4;        // 0..15
    const int sq   = tid & 15;        // float4 slot within row
    const float* gsrc = embed + (size_t)srow * DIM + sq * 4;
    const int ntiles = n_codes / TILE_N;

#if USE_ASYNC_LDS
    // Prime the double buffer: tiles 0 and 1 in flight (2 DMA ops each).
#pragma unroll
    for (int t = 0; t < 2; ++t) {
        const float* g = gsrc + (size_t)t * TILE_N * DIM;
        async_tile_load_b128(g, &lds[t][srow * LDS_STRIDE + sq * 4]);
        async_tile_load_b128(g + (size_t)SUB_N * DIM,
                             &lds[t][(srow + SUB_N) * LDS_STRIDE + sq * 4]);
    }
#else
    float4 stage0 = *(const float4*)gsrc;                            // tile 0
    float4 stage1 = *(const float4*)(gsrc + (size_t)SUB_N * DIM);
#endif
    float esq0_s = esq[l16];          // tile 0 column norms (cols 0..15)
    float esq1_s = esq[SUB_N + l16];  // tile 0 column norms (cols 16..31)

    for (int nt = 0; nt < ntiles; ++nt) {
        const int cur = (NBUF == 2) ? (nt & 1) : 0;

#if USE_ASYNC_LDS
        // Current tile's 2 DMA ops must be done; next tile's may stay in flight.
        if (nt + 1 < ntiles) __builtin_amdgcn_s_wait_asynccnt(2);
        else                 __builtin_amdgcn_s_wait_asynccnt(0);
        __syncthreads();
#else
        *(float4*)&lds[0][srow * LDS_STRIDE + sq * 4]           = stage0;
        *(float4*)&lds[0][(srow + SUB_N) * LDS_STRIDE + sq * 4] = stage1;
        __syncthreads();
#endif

        const float nesq0 = -0.5f * esq0_s;
        const float nesq1 = -0.5f * esq1_s;
        // Pipeline next tile's esq (and, on the sync path, its data).
        {
            const int nn = (nt + 1 < ntiles) ? nt + 1 : nt;
            esq0_s = esq[nn * TILE_N + l16];
            esq1_s = esq[nn * TILE_N + SUB_N + l16];
#if !USE_ASYNC_LDS
            stage0 = *(const float4*)(gsrc + (size_t)nn * TILE_N * DIM);
            stage1 = *(const float4*)(gsrc + (size_t)nn * TILE_N * DIM
                                           + (size_t)SUB_N * DIM);
#endif
        }

        v8f c0, c1;
#pragma unroll
        for (int i = 0; i < 8; ++i) { c0[i] = nesq0; c1[i] = nesq1; }

        // B fragments: VGPR0 = {K=0 | K=2}, VGPR1 = {K=1 | K=3}, N = l16.
        // Two independent accumulator chains share each A fragment.
        const float* brow = &lds[cur][l16 * LDS_STRIDE + hi * 2];
#pragma unroll
        for (int s = 0; s < KSTEPS; ++s) {
            v2f b0 = *(const v2f*)(brow + 4 * s);
            v2f b1 = *(const v2f*)(brow + SUB_N * LDS_STRIDE + 4 * s);
            c0 = __builtin_amdgcn_wmma_f32_16x16x4_f32(
                     false, a[s], false, b0, (short)0, c0, false, false);
            c1 = __builtin_amdgcn_wmma_f32_16x16x4_f32(
                     false, a[s], false, b1, (short)0, c1, false, false);
        }

        // ---- running argmax; sub-tile 0 first keeps lowest index on ties ----
        const int cand0 = nt * TILE_N + l16;
#pragma unroll
        for (int i = 0; i < 8; ++i) {
            if (c0[i] > best[i]) { best[i] = c0[i]; bidx[i] = cand0; }
        }
#pragma unroll
        for (int i = 0; i < 8; ++i) {
            if (c1[i] > best[i]) { best[i] = c1[i]; bidx[i] = cand0 + SUB_N; }
        }
        __syncthreads();   // everyone done reading lds[cur]

#if USE_ASYNC_LDS
        if (nt + 2 < ntiles) {
            const float* g = gsrc + (size_t)(nt + 2) * TILE_N * DIM;
            async_tile_load_b128(g, &lds[cur][srow * LDS_STRIDE + sq * 4]);
            async_tile_load_b128(g + (size_t)SUB_N * DIM,
                                 &lds[cur][(srow + SUB_N) * LDS_STRIDE + sq * 4]);
        }
#endif
    }

    // ---- per-row reduction across the 16 lanes of each half, then output ----
    // C/D layout: VGPR i holds row (i + 8*hi), column = l16.
#pragma unroll
    for (int i = 0; i < 8; ++i) {
        float bd = best[i];
        int   bi = bidx[i];
#pragma unroll
        for (int off = 8; off > 0; off >>= 1) {   // xor 8/4/2/1 stays in 16-group
            float od = __shfl_xor(bd, off, 32);
            int   oi = __shfl_xor(bi, off, 32);
            if (od > bd || (od == bd && oi < bi)) { bd = od; bi = oi; }
        }
        const int rglob = r0 + i + 8 * hi;
        // gather winning codeword: 16 lanes x float4 = 64 floats
        float4 e = *(const float4*)(embed + (size_t)bi * DIM + l16 * 4);
        *(float4*)(out_q + (size_t)rglob * DIM + l16 * 4) = e;
        if (l16 == 0) out_idx[rglob] = (float)bi;
    }
}

// ---------------------------------------------------------------------------
extern "C" void kernel_launch(void* const* d_in, const int* in_sizes, int n_in,
                              void* d_out, int out_size, void* d_ws, size_t ws_size,
                              hipStream_t stream) {
    const float* x     = (const float*)d_in[0];   // [8,4096,64] f32
    const float* embed = (const float*)d_in[1];   // [4096,64]   f32

    const int nrows   = in_sizes[0] / DIM;        // 32768
    const int n_codes = in_sizes[1] / DIM;        // 4096

    float* esq     = (float*)d_ws;                        // [n_codes]
    float* out_q   = (float*)d_out;                       // [nrows*64]
    float* out_idx = (float*)d_out + (size_t)nrows * DIM; // [nrows]

    vq_esq_kernel<<<(n_codes + 255) / 256, 256, 0, stream>>>(embed, esq, n_codes);

    vq_argmax_kernel<<<nrows / ROWS_PER_BLK, WAVES_PER_BLK * 32, 0, stream>>>(
        x, embed, esq, out_q, out_idx, n_codes);
}